// YOLOLayer_71305047049040
// MI455X (gfx1250) — compile-verified
//
#include <hip/hip_runtime.h>
#include <math.h>

typedef float v2f __attribute__((ext_vector_type(2)));
typedef float v4f __attribute__((ext_vector_type(4)));
typedef float v8f __attribute__((ext_vector_type(8)));

#define CIN    1024
#define FDIM   20
#define PDIM   400           // 20*20
#define COUT   255           // 3*85
#define MPAD   256
#define NCH    85
#define KC     32            // K chunk staged through LDS
#define AS_STR 34            // LDS row stride (pad for bank-conflict-free frags)
#define BS_STR 34

__device__ __constant__ float c_anchor_w[3] = {116.0f / 32.0f, 156.0f / 32.0f, 373.0f / 32.0f};
__device__ __constant__ float c_anchor_h[3] = { 90.0f / 32.0f, 198.0f / 32.0f, 326.0f / 32.0f};

// hardware v_exp_f32 path; ~1e-6 rel error, plenty for sigmoid/exp epilogue
__device__ __forceinline__ float fast_exp(float x) { return __expf(x); }
__device__ __forceinline__ float fast_sigmoid(float x) {
    return 1.0f / (1.0f + __expf(-x));
}

__global__ __launch_bounds__(128) void yolo_head_wmma(
    const float* __restrict__ xin,   // [64][1024][400]
    const float* __restrict__ wgt,   // [255][1024]
    const float* __restrict__ bias,  // [255]
    float* __restrict__ out)         // [64][1200][85]
{
    __shared__ float As[MPAD * AS_STR];   // 256 rows (co) x 32 k, padded
    __shared__ float Bs[16 * BS_STR];     // 16 cols (p)  x 32 k, transposed

    const int n   = blockIdx.y;
    const int p0  = blockIdx.x * 16;
    const int t   = threadIdx.x;
    const int wid = t >> 5;        // wave 0..3, owns M rows [64*wid, 64*wid+63]
    const int lane = t & 31;
    const int lo  = lane & 15;
    const int hi  = lane >> 4;

    const float* xb = xin + (size_t)n * CIN * PDIM + p0;

    v8f acc0 = {}, acc1 = {}, acc2 = {}, acc3 = {};

    for (int k0 = 0; k0 < CIN; k0 += KC) {
        __syncthreads();
        // ---- stage A chunk: 256 rows x 32 k (float4 per thread x 16) ----
        #pragma unroll
        for (int i = 0; i < 16; ++i) {
            const int idx = t + 128 * i;        // 0..2047
            const int r   = idx >> 3;           // 0..255
            const int q   = idx & 7;            // float4 slot within 32 k
            v4f v = {0.0f, 0.0f, 0.0f, 0.0f};
            if (r < COUT)
                v = *(const v4f*)(wgt + (size_t)r * CIN + k0 + q * 4);
            float* dst = &As[r * AS_STR + q * 4];
            dst[0] = v.x; dst[1] = v.y; dst[2] = v.z; dst[3] = v.w;
        }
        // ---- stage B chunk: 32 k-rows x 16 p; one coalesced float4/thread ----
        {
            const int kk = t >> 2;              // 0..31
            const int qf = t & 3;               // float4 slot within 16 p
            const v4f v = __builtin_nontemporal_load(
                (const v4f*)(xb + (size_t)(k0 + kk) * PDIM + qf * 4));
            // transpose into Bs[p][k]
            Bs[(qf * 4 + 0) * BS_STR + kk] = v.x;
            Bs[(qf * 4 + 1) * BS_STR + kk] = v.y;
            Bs[(qf * 4 + 2) * BS_STR + kk] = v.z;
            Bs[(qf * 4 + 3) * BS_STR + kk] = v.w;
        }
        __syncthreads();

        // ---- 8 k-steps x 4 M-tiles of v_wmma_f32_16x16x4_f32 ----
        #pragma unroll
        for (int kk = 0; kk < KC; kk += 4) {
            const int ksel = kk + 2 * hi;       // per-lane K pair selector
            const v2f bfrag = *(const v2f*)&Bs[lo * BS_STR + ksel];
            const float* arow = &As[(64 * wid + lo) * AS_STR + ksel];
            const v2f a0 = *(const v2f*)(arow + 0 * 16 * AS_STR);
            const v2f a1 = *(const v2f*)(arow + 1 * 16 * AS_STR);
            const v2f a2 = *(const v2f*)(arow + 2 * 16 * AS_STR);
            const v2f a3 = *(const v2f*)(arow + 3 * 16 * AS_STR);
            acc0 = __builtin_amdgcn_wmma_f32_16x16x4_f32(false, a0, false, bfrag,
                                                         (short)0, acc0, false, false);
            acc1 = __builtin_amdgcn_wmma_f32_16x16x4_f32(false, a1, false, bfrag,
                                                         (short)0, acc1, false, false);
            acc2 = __builtin_amdgcn_wmma_f32_16x16x4_f32(false, a2, false, bfrag,
                                                         (short)0, acc2, false, false);
            acc3 = __builtin_amdgcn_wmma_f32_16x16x4_f32(false, a3, false, bfrag,
                                                         (short)0, acc3, false, false);
        }
    }

    // ---- YOLO epilogue + permuted scatter store ----
    const int p    = p0 + lo;            // spatial index 0..399 (N-dim of tile)
    const int gy   = p / FDIM;
    const int gx   = p - gy * FDIM;
    const float fx = (float)gx;
    const float fy = (float)gy;

    #pragma unroll
    for (int tt = 0; tt < 4; ++tt) {
        const v8f acc = (tt == 0) ? acc0 : (tt == 1) ? acc1 : (tt == 2) ? acc2 : acc3;
        #pragma unroll
        for (int i = 0; i < 8; ++i) {
            const int co = 64 * wid + 16 * tt + i + 8 * hi;  // output channel
            if (co >= COUT) continue;                         // padded row 255
            const float raw = acc[i] + bias[co];
            const int a = co / NCH;
            const int c = co - a * NCH;
            float val;
            if (c == 0)      val = (fast_sigmoid(raw) + fx) * 32.0f;
            else if (c == 1) val = (fast_sigmoid(raw) + fy) * 32.0f;
            else if (c == 2) val = fast_exp(raw) * c_anchor_w[a] * 32.0f;
            else if (c == 3) val = fast_exp(raw) * c_anchor_h[a] * 32.0f;
            else             val = fast_sigmoid(raw);
            __builtin_nontemporal_store(
                val, &out[(size_t)n * (3 * PDIM * NCH) + (size_t)(a * PDIM + p) * NCH + c]);
        }
    }
}

extern "C" void kernel_launch(void* const* d_in, const int* in_sizes, int n_in,
                              void* d_out, int out_size, void* d_ws, size_t ws_size,
                              hipStream_t stream) {
    (void)in_sizes; (void)n_in; (void)d_ws; (void)ws_size; (void)out_size;
    const float* xin  = (const float*)d_in[0];
    const float* wgt  = (const float*)d_in[1];
    const float* bias = (const float*)d_in[2];
    float* out = (float*)d_out;

    dim3 grid(PDIM / 16, 64);   // 25 p-tiles x 64 images; each WG does all 255 channels
    dim3 block(128);            // 4 waves, 4 accumulators each
    yolo_head_wmma<<<grid, block, 0, stream>>>(xin, wgt, bias, out);
}